// RelRankLoss_83820581749160
// MI455X (gfx1250) — compile-verified
//
#include <hip/hip_runtime.h>
#include <cstdint>
#include <cstddef>

// ---------------------------------------------------------------------------
// RelRankLoss for MI455X (gfx1250, wave32).
// Streaming segmented reduction + per-group k-th order statistic selection.
// Deterministic: all floating accumulation is done in 64-bit fixed point
// (scale 2^32) with integer atomics (order-independent).
// ---------------------------------------------------------------------------

#define G 256          // NUM_GROUPS
#define TPB 1024       // 32 wave32 waves per workgroup (fills a WGP)
#define NBLK 256
#define CAND_CAP_MAX 6144

#if defined(__gfx1250__) && __has_builtin(__builtin_amdgcn_global_load_async_to_lds_b32)
#define HAVE_ASYNC_LDS 1
#else
#define HAVE_ASYNC_LDS 0
#endif

// ---- JAX threefry2x32 (key = PRNGKey(42) -> (0, 42)), paired evaluation ----
// Element j gets x0-output, element j+half gets x1-output (jax iota split).
__device__ __forceinline__ void threefry_pair(uint32_t j, uint32_t half,
                                              uint32_t& o0, uint32_t& o1) {
  const uint32_t k0 = 0u, k1v = 42u, k2v = 0u ^ 42u ^ 0x1BD11BDAu;
  uint32_t x0 = j + k0;
  uint32_t x1 = (j + half) + k1v;
#define TF_R(r) { x0 += x1; x1 = (x1 << (r)) | (x1 >> (32 - (r))); x1 ^= x0; }
  TF_R(13) TF_R(15) TF_R(26) TF_R(6)   x0 += k1v; x1 += k2v + 1u;
  TF_R(17) TF_R(29) TF_R(16) TF_R(24)  x0 += k2v; x1 += k0  + 2u;
  TF_R(13) TF_R(15) TF_R(26) TF_R(6)   x0 += k0;  x1 += k1v + 3u;
  TF_R(17) TF_R(29) TF_R(16) TF_R(24)  x0 += k1v; x1 += k2v + 4u;
  TF_R(13) TF_R(15) TF_R(26) TF_R(6)   x0 += k2v; x1 += k0  + 5u;
#undef TF_R
  o0 = x0; o1 = x1;
}

// exp(x) in 2^32 fixed point (u64). Clamp keeps worst-case group sums < 2^61.
__device__ __forceinline__ unsigned long long fixexp(float x) {
  float v = fminf(expf(x), 4096.0f);
  return (unsigned long long)((double)v * 4294967296.0);
}

// Stage img_id rows for this block's lo/hi element ranges into LDS via the
// CDNA5 async global->LDS path (falls back to plain loads if unavailable).
__device__ __forceinline__ void stage_gids(const int* img, unsigned rowLo0,
                                           unsigned rowHi0, int nRows, int* s_gid) {
  int tid = (int)threadIdx.x;
  for (int i = tid; i < 2 * nRows; i += (int)blockDim.x) {
    const int* src = (i < nRows) ? (img + rowLo0 + i) : (img + rowHi0 + (i - nRows));
#if HAVE_ASYNC_LDS
    __builtin_amdgcn_global_load_async_to_lds_b32(
        (__attribute__((address_space(1))) int*)(uintptr_t)src,
        (__attribute__((address_space(3))) int*)(uint32_t)(uintptr_t)(s_gid + i),
        0, 0);
#else
    s_gid[i] = *src;
#endif
  }
#if HAVE_ASYNC_LDS
#if __has_builtin(__builtin_amdgcn_s_wait_asynccnt)
  __builtin_amdgcn_s_wait_asynccnt(0);
#else
  asm volatile("s_wait_asynccnt 0" ::: "memory");
#endif
#endif
}

// ---------------- K0: zero the small workspace region ----------------------
__global__ void rrl_k0_init(unsigned long long* ws64) {
  int i = (int)(blockIdx.x * blockDim.x + threadIdx.x);
  if (i < 4096) ws64[i] = 0ull;  // 32 KiB
}

// ---------------- K1: counts, pos exp-sum, top-nibble histogram ------------
__global__ __launch_bounds__(TPB)
void rrl_k1_stats(const float* __restrict__ inp, const int* __restrict__ rel,
                  const int* __restrict__ img,
                  unsigned long long* __restrict__ pos_fix,
                  unsigned* __restrict__ pos_cnt, unsigned* __restrict__ neg_cnt,
                  unsigned* __restrict__ hist, unsigned halfN, unsigned ppb) {
  __shared__ unsigned s_hist[G * 16];
  __shared__ unsigned s_pcnt[G], s_ncnt[G];
  __shared__ unsigned long long s_pfix[G];
  __shared__ int s_gid[TPB];  // ppb/128 lo rows + ppb/128 hi rows

  int tid = (int)threadIdx.x;
  unsigned blockStart = blockIdx.x * ppb;
  int nRows = (int)(ppb >> 7);  // C == 128

  for (int i = tid; i < G * 16; i += TPB) s_hist[i] = 0u;
  if (tid < G) { s_pcnt[tid] = 0u; s_ncnt[tid] = 0u; s_pfix[tid] = 0ull; }
  stage_gids(img, blockStart >> 7, (blockStart + halfN) >> 7, nRows, s_gid);
  __syncthreads();

  auto proc = [&](uint32_t bits, float x, int r, int g) {
    if (r > 0) {
      atomicAdd(&s_pcnt[g], 1u);
      atomicAdd(&s_pfix[g], fixexp(x));
    } else {
      atomicAdd(&s_ncnt[g], 1u);
      atomicAdd(&s_hist[(g << 4) + (int)(bits >> 28)], 1u);
    }
  };

  for (unsigned it = 0; it < ppb; it += TPB) {
    unsigned l = it + (unsigned)tid;       // local pair index
    unsigned j = blockStart + l;           // global pair index, element lo
    __builtin_prefetch(inp + j + TPB, 0, 0);
    __builtin_prefetch(inp + j + halfN + TPB, 0, 0);
    uint32_t b_lo, b_hi;
    threefry_pair(j, halfN, b_lo, b_hi);
    float x_lo = inp[j];          int r_lo = rel[j];
    float x_hi = inp[j + halfN];  int r_hi = rel[j + halfN];
    int g_lo = s_gid[l >> 7];
    int g_hi = s_gid[nRows + (int)(l >> 7)];
    proc(b_lo, x_lo, r_lo, g_lo);
    proc(b_hi, x_hi, r_hi, g_hi);
  }
  __syncthreads();

  for (int i = tid; i < G * 16; i += TPB)
    if (s_hist[i]) atomicAdd(&hist[i], s_hist[i]);
  if (tid < G) {
    if (s_pcnt[tid]) atomicAdd(&pos_cnt[tid], s_pcnt[tid]);
    if (s_ncnt[tid]) atomicAdd(&neg_cnt[tid], s_ncnt[tid]);
    if (s_pfix[tid]) atomicAdd(&pos_fix[tid], s_pfix[tid]);
  }
}

// ---------------- K2: pick nibble bucket holding the k-th smallest key -----
__global__ void rrl_k2_prefix(const unsigned* __restrict__ pos_cnt,
                              const unsigned* __restrict__ neg_cnt,
                              const unsigned* __restrict__ hist,
                              int* __restrict__ prefix_nib,
                              unsigned* __restrict__ krem) {
  int g = (int)threadIdx.x;
  unsigned k = pos_cnt[g], nn = neg_cnt[g];
  int p; unsigned kr = 0u;
  if (k == 0u) {
    p = -1;                 // no positives: select nothing
  } else if (k >= nn) {
    p = 16;                 // select every negative
  } else {
    p = 15;
    unsigned cum = 0u;
    for (int nib = 0; nib < 16; ++nib) {
      unsigned c = hist[(g << 4) + nib];
      if (cum + c >= k) { p = nib; kr = k - cum; break; }
      cum += c;
    }
  }
  prefix_nib[g] = p;
  krem[g] = kr;
}

// ---------------- K3: sum sure-selected negatives; collect boundary cands --
__global__ __launch_bounds__(TPB)
void rrl_k3_collect(const float* __restrict__ inp, const int* __restrict__ rel,
                    const int* __restrict__ img,
                    const int* __restrict__ prefix_nib,
                    unsigned long long* __restrict__ selneg_fix,
                    unsigned* __restrict__ cand_cnt,
                    unsigned long long* __restrict__ cand_key,
                    unsigned long long* __restrict__ cand_fix,
                    int cap, unsigned halfN, unsigned ppb) {
  __shared__ unsigned long long s_sneg[G];
  __shared__ int s_pref[G];
  __shared__ int s_gid[TPB];

  int tid = (int)threadIdx.x;
  unsigned blockStart = blockIdx.x * ppb;
  int nRows = (int)(ppb >> 7);

  if (tid < G) { s_sneg[tid] = 0ull; s_pref[tid] = prefix_nib[tid]; }
  stage_gids(img, blockStart >> 7, (blockStart + halfN) >> 7, nRows, s_gid);
  __syncthreads();

  auto proc = [&](uint32_t bits, unsigned e, float x, int r, int g) {
    if (r <= 0) {
      int nib = (int)(bits >> 28);
      int p = s_pref[g];
      if (nib < p) {
        atomicAdd(&s_sneg[g], fixexp(x));
      } else if (nib == p && cap > 0) {
        unsigned slot = atomicAdd(&cand_cnt[g], 1u);
        if (slot < (unsigned)cap) {
          size_t o = (size_t)g * (size_t)cap + slot;
          // 48-bit key: (u23 << 25) | flat_index  (matches lexsort tie-break)
          cand_key[o] = ((unsigned long long)(bits >> 9) << 25) |
                        (unsigned long long)e;
          cand_fix[o] = fixexp(x);
        }
      }
    }
  };

  for (unsigned it = 0; it < ppb; it += TPB) {
    unsigned l = it + (unsigned)tid;
    unsigned j = blockStart + l;
    __builtin_prefetch(inp + j + TPB, 0, 0);
    __builtin_prefetch(inp + j + halfN + TPB, 0, 0);
    uint32_t b_lo, b_hi;
    threefry_pair(j, halfN, b_lo, b_hi);
    float x_lo = inp[j];          int r_lo = rel[j];
    float x_hi = inp[j + halfN];  int r_hi = rel[j + halfN];
    int g_lo = s_gid[l >> 7];
    int g_hi = s_gid[nRows + (int)(l >> 7)];
    proc(b_lo, j, x_lo, r_lo, g_lo);
    proc(b_hi, j + halfN, x_hi, r_hi, g_hi);
  }
  __syncthreads();
  if (tid < G && s_sneg[tid]) atomicAdd(&selneg_fix[tid], s_sneg[tid]);
}

// ---------------- K4: exact k-th order statistic among candidates ----------
__global__ __launch_bounds__(256)
void rrl_k4_select(const int* __restrict__ prefix_nib,
                   const unsigned* __restrict__ krem_arr,
                   const unsigned* __restrict__ cand_cnt,
                   const unsigned long long* __restrict__ cand_key,
                   const unsigned long long* __restrict__ cand_fix,
                   unsigned long long* __restrict__ selneg_fix, int cap) {
  int g = (int)blockIdx.x;
  int tid = (int)threadIdx.x;
  int p = prefix_nib[g];
  if (p < 0 || p > 15) return;              // nothing / everything: done in K3

  __shared__ unsigned s_hcnt[256];
  __shared__ unsigned long long s_hfix[256];
  __shared__ unsigned long long s_prefv;
  __shared__ unsigned s_kr;
  __shared__ unsigned long long s_sum;

  unsigned m = cand_cnt[g];
  if (m > (unsigned)cap) m = (unsigned)cap;
  if (tid == 0) { s_prefv = 0ull; s_kr = krem_arr[g]; s_sum = 0ull; }
  __syncthreads();
  if (s_kr == 0u) return;                   // uniform across block

  const unsigned long long* keys = cand_key + (size_t)g * (size_t)cap;
  const unsigned long long* fixs = cand_fix + (size_t)g * (size_t)cap;

  for (int round = 0; round < 6; ++round) {
    int shift = 40 - 8 * round;             // 48-bit key, 8-bit digits
    for (int i = tid; i < 256; i += 256) { s_hcnt[i] = 0u; s_hfix[i] = 0ull; }
    __syncthreads();
    unsigned long long pref = s_prefv;
    for (unsigned i = (unsigned)tid; i < m; i += 256u) {
      unsigned long long k = keys[i];
      if ((k >> (shift + 8)) == pref) {
        unsigned d = (unsigned)(k >> shift) & 255u;
        atomicAdd(&s_hcnt[d], 1u);
        atomicAdd(&s_hfix[d], fixs[i]);
      }
    }
    __syncthreads();
    if (tid == 0) {
      unsigned cum = 0u;
      unsigned long long below = 0ull;
      int chosen = 255;
      for (int d = 0; d < 256; ++d) {
        if (cum + s_hcnt[d] >= s_kr) { chosen = d; break; }
        cum += s_hcnt[d]; below += s_hfix[d];
      }
      s_sum += below;
      s_kr -= cum;
      s_prefv = (s_prefv << 8) | (unsigned long long)chosen;
      if (round == 5) s_sum += s_hfix[chosen];  // keys are unique: exactly 1
    }
    __syncthreads();
  }
  if (tid == 0) selneg_fix[g] += s_sum;     // sole writer at this stage
}

// ---------------- K5: per-group loss + final mean --------------------------
__global__ void rrl_k5_loss(const unsigned* __restrict__ pos_cnt,
                            const unsigned* __restrict__ neg_cnt,
                            const unsigned long long* __restrict__ pos_fix,
                            const unsigned long long* __restrict__ selneg_fix,
                            float* __restrict__ out) {
  __shared__ double s_loss[G];
  __shared__ int s_pres[G];
  int g = (int)threadIdx.x;
  unsigned pc = pos_cnt[g], nc = neg_cnt[g];
  bool present = (pc + nc) > 0u;
  bool valid = pc > 0u;
  double lg = 0.0;
  if (valid) {
    const double inv = 1.0 / 4294967296.0;
    double ps = (double)pos_fix[g] * inv;
    double bs = (double)(pos_fix[g] + selneg_fix[g]) * inv;
    lg = log(bs) - log(ps);                 // == -log(pos/both)
  }
  s_loss[g] = present ? lg : 0.0;
  s_pres[g] = present ? 1 : 0;
  __syncthreads();
  for (int s = G / 2; s > 0; s >>= 1) {
    if (g < s) { s_loss[g] += s_loss[g + s]; s_pres[g] += s_pres[g + s]; }
    __syncthreads();
  }
  if (g == 0) {
    double np = (s_pres[0] > 0) ? (double)s_pres[0] : 1.0;
    out[0] = (float)(s_loss[0] / np);
  }
}

// ---------------------------------------------------------------------------
extern "C" void kernel_launch(void* const* d_in, const int* in_sizes, int n_in,
                              void* d_out, int out_size, void* d_ws, size_t ws_size,
                              hipStream_t stream) {
  const float* inp = (const float*)d_in[0];
  const int*   rel = (const int*)d_in[1];
  const int*   img = (const int*)d_in[2];

  unsigned NC    = (unsigned)in_sizes[0];   // 33554432 (N=262144, C=128)
  unsigned halfN = NC >> 1;                 // threefry iota split point
  unsigned ppb   = halfN / NBLK;            // pairs per block (65536)

  char* ws = (char*)d_ws;
  unsigned long long* pos_fix = (unsigned long long*)(ws + 0);
  unsigned long long* selneg  = (unsigned long long*)(ws + 2048);
  unsigned* pos_cnt = (unsigned*)(ws + 4096);
  unsigned* neg_cnt = (unsigned*)(ws + 5120);
  unsigned* hist    = (unsigned*)(ws + 6144);       // G*16 u32
  int*      prefix  = (int*)(ws + 22528);
  unsigned* krem    = (unsigned*)(ws + 23552);
  unsigned* ccnt    = (unsigned*)(ws + 24576);
  const size_t cand_base = 32768;

  size_t avail = (ws_size > cand_base) ? (ws_size - cand_base) : 0;
  size_t capz  = avail / ((size_t)G * 16u);         // key + fix, 8B each
  int cap = (int)((capz > CAND_CAP_MAX) ? CAND_CAP_MAX : capz);
  unsigned long long* ckey = (unsigned long long*)(ws + cand_base);
  unsigned long long* cfix = ckey + (size_t)G * (size_t)cap;

  rrl_k0_init<<<4, 1024, 0, stream>>>((unsigned long long*)ws);
  rrl_k1_stats<<<NBLK, TPB, 0, stream>>>(inp, rel, img, pos_fix, pos_cnt,
                                         neg_cnt, hist, halfN, ppb);
  rrl_k2_prefix<<<1, G, 0, stream>>>(pos_cnt, neg_cnt, hist, prefix, krem);
  rrl_k3_collect<<<NBLK, TPB, 0, stream>>>(inp, rel, img, prefix, selneg, ccnt,
                                           ckey, cfix, cap, halfN, ppb);
  rrl_k4_select<<<G, 256, 0, stream>>>(prefix, krem, ccnt, ckey, cfix, selneg, cap);
  rrl_k5_loss<<<1, G, 0, stream>>>(pos_cnt, neg_cnt, pos_fix, selneg,
                                   (float*)d_out);
}